// ChamferLoss_6262062318051
// MI455X (gfx1250) — compile-verified
//
#include <hip/hip_runtime.h>

// ---------------------------------------------------------------------------
// Chamfer loss on MI455X (gfx1250), wave32, WMMA f32 16x16x4.
//
// dist(q, r) = |q|^2 + |r|^2 - 2 q.r
// Per 16x16 tile:  D = A x B  with
//   A row m = (qx, qy, qz, 1)            (K padded 3 -> 4)
//   B col n = (-2rx, -2ry, -2rz, |r|^2)
// so D[m][n] = -2 q.r + |r|^2; the row-constant |q|^2 is folded into the
// final scalar sum (sum of min-dists = sum(rowmin) + sum(|q|^2)).
//
// Refs are pre-swizzled into WMMA B-register layout (float2 per (tile,lane)),
// staged through a DOUBLE-BUFFERED 2x32KB LDS window with CDNA5 async
// global->LDS loads (ASYNCcnt) overlapping compute, and the inner loop is
// software-pipelined: preload next B tile, 2 WMMA per m-tile pair, min fold.
// ---------------------------------------------------------------------------

typedef __attribute__((ext_vector_type(2))) float v2f;
typedef __attribute__((ext_vector_type(8))) float v8f;
typedef __attribute__((ext_vector_type(4))) int   v4i;

#define NPTS  8192
#define B_SZ  8
#define CHUNK 2048             // ref points per LDS buffer (32 KB)
#define NCHUNK (NPTS / CHUNK)
#define NTILE (CHUNK / 16)     // 128 n-tiles per chunk
#define NWAVE 8                // waves per workgroup (256 threads, wave32)
#define ROWS_PER_WAVE 32       // two 16-row m-tiles per wave
#define MBLK  (NWAVE * ROWS_PER_WAVE)   // 256 query rows per workgroup
#define FLT_BIG 3.4e38f

#if defined(__gfx1250__) &&                                            \
    __has_builtin(__builtin_amdgcn_global_load_async_to_lds_b128) &&   \
    __has_builtin(__builtin_amdgcn_s_wait_asynccnt)
#define USE_ASYNC_LDS 1
#else
#define USE_ASYNC_LDS 0
#endif

// Build B-layout ref arrays: per point i (tile=i/16, slot=i%16):
//   refB2[tile*32 + slot     ] = (-2x, -2y)   (K0,K1 lane)
//   refB2[tile*32 + slot + 16] = (-2z, |p|^2) (K2,K3 lane)
__global__ __launch_bounds__(256) void chamfer_aug_kernel(
    const float* __restrict__ preds, const float* __restrict__ gts,
    v2f* __restrict__ refBPreds, v2f* __restrict__ refBGts) {
  int t = blockIdx.x * 256 + threadIdx.x;     // 0 .. 2*B*NPTS-1
  const int n = B_SZ * NPTS;
  const float* src = (t < n) ? preds : gts;
  v2f* dst         = (t < n) ? refBPreds : refBGts;
  int i = (t < n) ? t : (t - n);
  float x = src[3 * i + 0];
  float y = src[3 * i + 1];
  float z = src[3 * i + 2];
  int tile = i >> 4;
  int slot = i & 15;
  v2f lo2, hi2;
  lo2.x = -2.0f * x;
  lo2.y = -2.0f * y;
  hi2.x = -2.0f * z;
  hi2.y = x * x + y * y + z * z;
  dst[tile * 32 + slot]      = lo2;
  dst[tile * 32 + slot + 16] = hi2;
}

// One directed Chamfer pass. grid = B_SZ * (NPTS / MBLK) = 256 blocks.
// partials[blk] = sum over the block's 256 query rows of
//                 (min_n(-2 q.r + |r|^2) + |q|^2).
__global__ __launch_bounds__(256) void chamfer_dir_kernel(
    const float* __restrict__ query,      // raw [B, NPTS, 3]
    const v2f* __restrict__ refB,         // B-layout [B, NPTS/16, 32]
    float* __restrict__ partials) {
  // Double-buffered B-layout ref window: 2 x 32 KB = 64 KB.
  __shared__ alignas(16) v2f sref[2][CHUNK * 2];

  const int blk  = blockIdx.x;
  const int b    = blk / (NPTS / MBLK);
  const int mblk = blk % (NPTS / MBLK);
  const int tid  = threadIdx.x;
  const int w    = tid >> 5;
  const int lane = tid & 31;
  const bool lo  = (lane < 16);

  const float* q = query + (size_t)b * NPTS * 3;
  // float4 view of this batch's B-layout refs (1 float4 per point, permuted)
  const float4* r4 = (const float4*)(refB + (size_t)b * NPTS * 2);

  // Stage one chunk into LDS buffer `buf` (async on CDNA5, ASYNCcnt-tracked).
  auto stage = [&](int buf, int c0) {
    float4* db = (float4*)sref[buf];
#pragma unroll
    for (int i = tid; i < CHUNK; i += 256) {
#if USE_ASYNC_LDS
      __builtin_amdgcn_global_load_async_to_lds_b128(
          (__attribute__((address_space(1))) v4i*)&r4[c0 + i],
          (__attribute__((address_space(3))) v4i*)&db[i],
          /*offset=*/0, /*cpol=*/0);
#else
      db[i] = r4[c0 + i];
      __builtin_prefetch(&r4[c0 + CHUNK + i], 0, 1);
#endif
    }
  };

  // --- two A matrices per wave (rows m0..m0+15 and m0+16..m0+31) ---
  const int m0   = mblk * MBLK + w * ROWS_PER_WAVE;
  const int rowA = m0 + (lane & 15);
  const int rowB = rowA + 16;
  v2f a0, a1;
  a0.x = lo ? q[3 * rowA + 0] : q[3 * rowA + 2];
  a0.y = lo ? q[3 * rowA + 1] : 1.0f;
  a1.x = lo ? q[3 * rowB + 0] : q[3 * rowB + 2];
  a1.y = lo ? q[3 * rowB + 1] : 1.0f;
  // per-lane |q|^2 contributions (exclude the augmented 1):
  float sq = a0.x * a0.x + a1.x * a1.x +
             (lo ? (a0.y * a0.y + a1.y * a1.y) : 0.0f);

  v8f acc0, acc1;
#pragma unroll
  for (int i = 0; i < 8; ++i) { acc0[i] = FLT_BIG; acc1[i] = FLT_BIG; }

  stage(0, 0);
#if USE_ASYNC_LDS
  __builtin_amdgcn_s_wait_asynccnt(0);
#endif
  __syncthreads();

  for (int c = 0; c < NCHUNK; ++c) {
    const int buf = c & 1;
    // Kick off async staging of the NEXT chunk into the other buffer while
    // this chunk is consumed (its previous readers passed the last barrier).
    if (c + 1 < NCHUNK) stage(buf ^ 1, (c + 1) * CHUNK);

    const v2f* sb = sref[buf];
    // Software pipeline: preload next tile's B column pair before the mins.
    v2f bm = sb[lane];
#pragma unroll 2
    for (int t = 0; t < NTILE; ++t) {
      v2f bmn = sb[(((t + 1) & (NTILE - 1)) << 5) + lane];
      v8f cz = {};
      v8f d0 = __builtin_amdgcn_wmma_f32_16x16x4_f32(
          false, a0, false, bm, (short)0, cz, false, false);
      v8f d1 = __builtin_amdgcn_wmma_f32_16x16x4_f32(
          false, a1, false, bm, (short)0, cz, false, false);
#pragma unroll
      for (int i = 0; i < 8; ++i) {
        acc0[i] = fminf(acc0[i], d0[i]);
        acc1[i] = fminf(acc1[i], d1[i]);
      }
      bm = bmn;
    }

#if USE_ASYNC_LDS
    __builtin_amdgcn_s_wait_asynccnt(0);   // next buffer fully landed in LDS
#endif
    __syncthreads();
  }

  // ---- row-min: butterfly min across the 16 lanes of each half ----
  // C/D layout: VGPR i of acc0 -> row m0+i (lanes 0-15), m0+8+i (lanes 16-31);
  // acc1 -> rows m0+16+i / m0+24+i.
  float rowsum = 0.0f;
#pragma unroll
  for (int i = 0; i < 8; ++i) {
    float v0 = acc0[i];
    v0 = fminf(v0, __shfl_xor(v0, 1, 32));
    v0 = fminf(v0, __shfl_xor(v0, 2, 32));
    v0 = fminf(v0, __shfl_xor(v0, 4, 32));
    v0 = fminf(v0, __shfl_xor(v0, 8, 32));
    float v1 = acc1[i];
    v1 = fminf(v1, __shfl_xor(v1, 1, 32));
    v1 = fminf(v1, __shfl_xor(v1, 2, 32));
    v1 = fminf(v1, __shfl_xor(v1, 4, 32));
    v1 = fminf(v1, __shfl_xor(v1, 8, 32));
    rowsum += v0 + v1;
  }
  // lanes 0 and 16 together cover each of the 32 rows exactly once.
  float contrib = ((lane == 0) || (lane == 16)) ? rowsum : 0.0f;
  contrib += sq;
#pragma unroll
  for (int m = 1; m < 32; m <<= 1) contrib += __shfl_xor(contrib, m, 32);

  // Per-wave partials: alias over the (now dead) LDS buffer 0.
  float* wsum = (float*)&sref[0][0];
  if (lane == 0) wsum[w] = contrib;
  __syncthreads();
  if (tid == 0) {
    float s = 0.0f;
#pragma unroll
    for (int i = 0; i < NWAVE; ++i) s += wsum[i];
    partials[blk] = s;
  }
}

// Final fixed-order reduction: out = (S1 + S2) / (B*NPTS).
__global__ __launch_bounds__(256) void chamfer_reduce_kernel(
    const float* __restrict__ p1, const float* __restrict__ p2,
    float* __restrict__ out, int nparts) {
  __shared__ float s[256];
  int tid = threadIdx.x;
  float v = 0.0f;
  for (int i = tid; i < nparts; i += 256) v += p1[i] + p2[i];
  s[tid] = v;
  __syncthreads();
  for (int st = 128; st > 0; st >>= 1) {
    if (tid < st) s[tid] += s[tid + st];
    __syncthreads();
  }
  if (tid == 0) out[0] = s[0] * (1.0f / (float)(B_SZ * NPTS));
}

extern "C" void kernel_launch(void* const* d_in, const int* in_sizes, int n_in,
                              void* d_out, int out_size, void* d_ws, size_t ws_size,
                              hipStream_t stream) {
  const float* preds = (const float*)d_in[0];   // [8, 8192, 3] f32
  const float* gts   = (const float*)d_in[1];   // [8, 8192, 3] f32
  float* out = (float*)d_out;                   // scalar f32

  // Workspace: refBPreds (1 MB) | refBGts (1 MB) | part1 | part2
  char* ws = (char*)d_ws;
  v2f*   refBPreds = (v2f*)(ws);
  v2f*   refBGts   = (v2f*)(ws + (size_t)(1 << 20));
  float* part1     = (float*)(ws + (size_t)(2 << 20));
  float* part2     = (float*)(ws + (size_t)(2 << 20) + 4096);

  const int nAug    = 2 * B_SZ * NPTS;          // 131072 points total
  const int nBlocks = B_SZ * (NPTS / MBLK);     // 256 per direction

  chamfer_aug_kernel<<<nAug / 256, 256, 0, stream>>>(preds, gts, refBPreds, refBGts);
  // loss_1: for each pred, nearest gt  (query = preds, ref = gts)
  chamfer_dir_kernel<<<nBlocks, 256, 0, stream>>>(preds, refBGts, part1);
  // loss_2: for each gt, nearest pred  (query = gts, ref = preds)
  chamfer_dir_kernel<<<nBlocks, 256, 0, stream>>>(gts, refBPreds, part2);
  chamfer_reduce_kernel<<<1, 256, 0, stream>>>(part1, part2, out, nBlocks);
}